// SoftDTW_3058016714975
// MI455X (gfx1250) — compile-verified
//
#include <hip/hip_runtime.h>
#include <hip/hip_bf16.h>

// ---------------------------------------------------------------------------
// Soft-DTW forward: B=32, S=512, D=64, gamma=1, no band.
// Phase 1: D[b,i,j] = x2[i] + y2[j] - 2 * X[b,i]·Y[b,j]  via V_WMMA_F32_16X16X4_F32
// Phase 2: anti-diagonal wavefront DP, one workgroup per batch, LDS diagonals.
// ---------------------------------------------------------------------------

#define BATCH 32
#define SLEN  512
#define FDIM  64
#define NTILE (SLEN / 16)                 // 32 tiles per dim
#define TILES_PER_BATCH (NTILE * NTILE)   // 1024

typedef __attribute__((ext_vector_type(2))) float v2f;
typedef __attribute__((ext_vector_type(8))) float v8f;

// ---------------------------------------------------------------------------
// Kernel 0: row norms x2[b,i] = sum_k X[b,i,k]^2   (and same for Y)
// one thread per (b, row); float4 loads (16 iterations over 64 features)
// ---------------------------------------------------------------------------
__global__ void sdtw_norms_kernel(const float* __restrict__ X,
                                  const float* __restrict__ Y,
                                  float* __restrict__ x2,
                                  float* __restrict__ y2) {
    int r = blockIdx.x * blockDim.x + threadIdx.x;   // 0 .. BATCH*SLEN-1
    if (r >= BATCH * SLEN) return;
    const float4* xp = (const float4*)(X + (size_t)r * FDIM);
    const float4* yp = (const float4*)(Y + (size_t)r * FDIM);
    float sx = 0.0f, sy = 0.0f;
#pragma unroll
    for (int k = 0; k < FDIM / 4; ++k) {
        float4 a = xp[k];
        float4 b = yp[k];
        sx += a.x * a.x + a.y * a.y + a.z * a.z + a.w * a.w;
        sy += b.x * b.x + b.y * b.y + b.z * b.z + b.w * b.w;
    }
    x2[r] = sx;
    y2[r] = sy;
}

// ---------------------------------------------------------------------------
// Kernel 1: one wave32 per 16x16 output tile of D[b] = x2 + y2 - 2*X Y^T.
// V_WMMA_F32_16X16X4_F32 accumulated over K=64 in 16 steps.
//
// f32 16x16x4 A layout (ISA 7.12.2): lanes 0-15 -> rows M=0..15 with K=0 (v0),
// K=1 (v1); lanes 16-31 -> same rows with K=2 (v0), K=3 (v1). B mirrors with
// N in place of M (B[k,n] = Y[n,k] since we multiply by Y^T).
// C/D layout: VGPR r, lanes 0-15 -> M=r, lanes 16-31 -> M=8+r; N = lane&15.
// ---------------------------------------------------------------------------
__global__ void sdtw_dist_wmma_kernel(const float* __restrict__ X,
                                      const float* __restrict__ Y,
                                      const float* __restrict__ x2,
                                      const float* __restrict__ y2,
                                      float* __restrict__ Dout) {
    const int wavesPerBlock = blockDim.x >> 5;
    const int wave = blockIdx.x * wavesPerBlock + (threadIdx.x >> 5);
    const int lane = threadIdx.x & 31;

    const int b  = wave / TILES_PER_BATCH;
    const int t  = wave % TILES_PER_BATCH;
    const int bi = t / NTILE;              // row tile
    const int bj = t % NTILE;              // col tile

    const int half = lane >> 4;            // 0: K={0,1}, 1: K={2,3}
    const int idx  = lane & 15;            // M (for A) / N (for B)

    const float* Xrow = X + ((size_t)b * SLEN + (size_t)(bi * 16 + idx)) * FDIM;
    const float* Yrow = Y + ((size_t)b * SLEN + (size_t)(bj * 16 + idx)) * FDIM;

    v8f c = {};
#pragma unroll
    for (int k0 = 0; k0 < FDIM; k0 += 4) {
        const int k = k0 + half * 2;                    // even -> 8B aligned
        v2f a  = *(const v2f*)(Xrow + k);               // A: 16x4 slice of X
        v2f bm = *(const v2f*)(Yrow + k);               // B: 4x16 slice of Y^T
        // 8 args: (neg_a, A, neg_b, B, c_mod, C, reuse_a, reuse_b)
        c = __builtin_amdgcn_wmma_f32_16x16x4_f32(
                false, a, false, bm, (short)0, c, false, false);
    }

    const int col = bj * 16 + idx;
    const float yn = y2[(size_t)b * SLEN + col];
    float* Db = Dout + (size_t)b * SLEN * SLEN;
#pragma unroll
    for (int r = 0; r < 8; ++r) {
        const int row = bi * 16 + half * 8 + r;
        const float xn = x2[(size_t)b * SLEN + row];
        Db[(size_t)row * SLEN + col] = xn + yn - 2.0f * c[r];
    }
}

// ---------------------------------------------------------------------------
// Kernel 2: wavefront soft-DTW. One block per batch, 512 threads = 16 waves.
// Thread i owns row i; diagonal p visits cell (i, p-i). Three rotating LDS
// diagonal buffers -> exactly one barrier per diagonal step.
// softmin_gamma=1(a,b,c) = -log(e^-a + e^-b + e^-c), max-stabilized.
// ---------------------------------------------------------------------------
__global__ void __launch_bounds__(SLEN)
sdtw_dp_kernel(const float* __restrict__ Dmat, float* __restrict__ out) {
    const int b = blockIdx.x;
    const int i = threadIdx.x;            // row 0..SLEN-1
    const float INF = __builtin_inff();

    __shared__ float R[3][SLEN];
    R[0][i] = INF;
    R[1][i] = INF;
    R[2][i] = INF;
    __syncthreads();

    const float* __restrict__ Drow =
        Dmat + (size_t)b * SLEN * SLEN + (size_t)i * SLEN;

    for (int p = 0; p < 2 * SLEN - 1; ++p) {
        const float* r2 = R[(p + 1) % 3];   // diagonal p-2
        const float* r1 = R[(p + 2) % 3];   // diagonal p-1
        float*       rc = R[p % 3];         // diagonal p (write)

        const int j = p - i;
        float nv = INF;
        if (j >= 0 && j < SLEN) {
            // diag / up / left neighbors (inf where the neighbor is invalid)
            const float dd = (i == 0) ? ((p == 0) ? 0.0f : INF) : r2[i - 1];
            const float up = (i == 0) ? INF : r1[i - 1];
            const float lf = r1[i];

            const float m = fmaxf(fmaxf(-dd, -up), -lf);
            float sm;
            if (m == -INF) {
                sm = INF;
            } else {
                const float s = __expf(-dd - m) + __expf(-up - m) + __expf(-lf - m);
                sm = -(__logf(s) + m);
            }
            // stream the distance row; prefetch a cacheline ahead for this row
            __builtin_prefetch(Drow + j + 32, 0, 0);
            nv = Drow[j] + sm;
        }
        rc[i] = nv;
        __syncthreads();
    }

    // final diagonal p = 2S-2 lives in R[(2S-2)%3]; answer is cell (S-1,S-1)
    if (i == SLEN - 1) {
        out[b] = R[(2 * SLEN - 2) % 3][SLEN - 1];
    }
}

// ---------------------------------------------------------------------------
// Launch: norms -> WMMA distance matrix -> wavefront DP, all on `stream`.
// Workspace: [ x2 : 16384 f32 | y2 : 16384 f32 | D : 32*512*512 f32 (32 MB) ]
// ---------------------------------------------------------------------------
extern "C" void kernel_launch(void* const* d_in, const int* in_sizes, int n_in,
                              void* d_out, int out_size, void* d_ws, size_t ws_size,
                              hipStream_t stream) {
    const float* X = (const float*)d_in[0];
    const float* Y = (const float*)d_in[1];
    float* ws  = (float*)d_ws;
    float* x2  = ws;                          // BATCH*SLEN
    float* y2  = ws + BATCH * SLEN;           // BATCH*SLEN
    float* Dm  = ws + 2 * BATCH * SLEN;       // BATCH*SLEN*SLEN
    float* out = (float*)d_out;               // BATCH

    // 16384 rows, 256 threads/block
    sdtw_norms_kernel<<<(BATCH * SLEN) / 256, 256, 0, stream>>>(X, Y, x2, y2);

    // 32768 tiles total, 8 waves (256 threads) per block -> 4096 blocks
    sdtw_dist_wmma_kernel<<<(BATCH * TILES_PER_BATCH) / 8, 256, 0, stream>>>(
        X, Y, x2, y2, Dm);

    // one block per batch, 512 threads (16 wave32s)
    sdtw_dp_kernel<<<BATCH, SLEN, 0, stream>>>(Dm, out);
}